// SPDMLIFConstraint_18717467476431
// MI455X (gfx1250) — compile-verified
//
#include <hip/hip_runtime.h>
#include <math.h>
#include <stdint.h>

typedef __attribute__((ext_vector_type(2))) float v2f;
typedef __attribute__((ext_vector_type(8))) float v8f;

#define B_   4
#define T_   2048
#define D_   1024
#define F_   4096
#define FT   64     // F columns per workgroup
#define NT   64     // number of f tiles (F_/FT)
#define TT   32     // T rows per block iteration
#define KC   64     // K chunk staged in LDS
#define NKC  (D_ / KC)
#define AST  68     // padded LDS row stride for A tile (floats)
#define WST  68     // padded LDS row stride for W tile (floats)
#define GST  65     // padded LDS row stride for G tile (floats)

// CDNA5 async memory->LDS copy (16B per lane), tracked by ASYNCcnt.
__device__ __forceinline__ void async_copy16(uint32_t lds_off, const float* gptr) {
    asm volatile("global_load_async_to_lds_b128 %0, %1, off"
                 :: "v"(lds_off), "v"(gptr)
                 : "memory");
}
__device__ __forceinline__ void wait_async0() {
    asm volatile("s_wait_asynccnt 0x0" ::: "memory");
}

// Issue the 6 async b128 copies that stage one (A 32xKC, W 64xKC) chunk.
__device__ __forceinline__ void stage_chunk(
    float* AldsBuf, float* WldsBuf,
    const float* __restrict__ Ab, const float* __restrict__ W,
    int f_base, int t0, int kc, int tid)
{
    // A chunk: 32 x 64 floats = 512 float4, 2 per thread
    int idx = tid;
    int r = idx >> 4, c = (idx & 15) << 2;
    async_copy16((uint32_t)(uintptr_t)&AldsBuf[r * AST + c],
                 Ab + (size_t)(t0 + r) * D_ + kc + c);
    idx = tid + 256; r = idx >> 4; c = (idx & 15) << 2;
    async_copy16((uint32_t)(uintptr_t)&AldsBuf[r * AST + c],
                 Ab + (size_t)(t0 + r) * D_ + kc + c);
    // W chunk: 64 x 64 floats = 1024 float4, 4 per thread
    #pragma unroll
    for (int i = 0; i < 4; ++i) {
        int idx2 = tid + i * 256;
        int r2 = idx2 >> 4, c2 = (idx2 & 15) << 2;
        async_copy16((uint32_t)(uintptr_t)&WldsBuf[r2 * WST + c2],
                     W + (size_t)(f_base + r2) * D_ + kc + c2);
    }
}

// Fused: gate = A @ W^T (fp32 WMMA), LIF scan over T, per-(b,t) partial sums
// of {spike count, spike·act, act^2} over this workgroup's 64 f columns.
__global__ __launch_bounds__(256) void lif_gemm_kernel(
    const float* __restrict__ A,     // mlp_input  (B,T,D)
    const float* __restrict__ act,   // mlp_activation (B,T,F)
    const float* __restrict__ W,     // W_proj (F,D)
    const float* __restrict__ theta, // (F,)
    float* __restrict__ cntp,        // (B*NT, T) partial spike counts
    float* __restrict__ dotp,        // (B*NT, T) partial spike·act
    float* __restrict__ ssp)         // (B*NT, T) partial act^2
{
    __shared__ __align__(16) float Alds[2][TT * AST];
    __shared__ __align__(16) float Wlds[2][FT * WST];
    __shared__ __align__(16) float Glds[TT * GST];

    const int tid  = threadIdx.x;
    const int wgid = blockIdx.x;
    const int b    = wgid >> 6;          // 4 b's
    const int tile = wgid & (NT - 1);    // 64 f tiles
    const int f_base = tile * FT;

    const int wave = tid >> 5;           // 8 waves (wave32)
    const int lane = tid & 31;
    const int hi   = lane >> 4;          // 0/1: which K-pair / M-half
    const int l16  = lane & 15;
    const int mi   = wave & 1;           // M subtile (rows 16*mi..)
    const int ni   = wave >> 1;          // N subtile (cols 16*ni..)

    const float* Ab   = A   + (size_t)b * T_ * D_;
    const float* actb = act + (size_t)b * T_ * F_;

    // LIF state: threads 0..63 each own one f column
    float mem = 0.0f;
    float th  = 0.0f;
    if (tid < FT) th = theta[f_base + tid];

    const int pbase = (b * NT + tile) * T_;

    // Prologue: async-stage chunk 0 of t-block 0 into buffer 0.
    stage_chunk(Alds[0], Wlds[0], Ab, W, f_base, 0, 0, tid);

    for (int tb = 0; tb < T_ / TT; ++tb) {
        const int t0 = tb * TT;
        v8f acc = {0.f, 0.f, 0.f, 0.f, 0.f, 0.f, 0.f, 0.f};

        for (int j = 0; j < NKC; ++j) {
            // Current chunk's async copies (issued last iteration) must land.
            wait_async0();
            __syncthreads();

            // Issue next chunk (or chunk 0 of the next t-block) into the
            // other buffer; overlaps with the WMMA stream below (and, for
            // the cross-t-block issue, with the scan/reduce tail).
            if (j + 1 < NKC) {
                stage_chunk(Alds[(j + 1) & 1], Wlds[(j + 1) & 1],
                            Ab, W, f_base, t0, (j + 1) * KC, tid);
            } else if (tb + 1 < T_ / TT) {
                stage_chunk(Alds[0], Wlds[0], Ab, W, f_base, t0 + TT, 0, tid);
            }

            // fp32 WMMA K-loop on buffer j&1.
            // A frag lane: M=l16, K = kk+2*hi+{0,1}
            // B frag lane: N=l16, K = kk+2*hi+{0,1}  (B = W^T)
            const float* arow = &Alds[j & 1][(mi * 16 + l16) * AST];
            const float* brow = &Wlds[j & 1][(ni * 16 + l16) * WST];
            #pragma unroll
            for (int kk = 0; kk < KC; kk += 4) {
                v2f av = *(const v2f*)(arow + kk + 2 * hi);
                v2f bv = *(const v2f*)(brow + kk + 2 * hi);
                acc = __builtin_amdgcn_wmma_f32_16x16x4_f32(
                    false, av, false, bv, (short)0, acc, false, false);
            }
        }
        __syncthreads();   // all waves done reading A/W LDS for this t-block

        // Spill G tile (32 x 64) to LDS. C/D layout: VGPR v, lanes0-15: M=v,
        // lanes16-31: M=v+8; N = lane&15.
        #pragma unroll
        for (int v = 0; v < 8; ++v) {
            int m = mi * 16 + v + 8 * hi;
            int n = ni * 16 + l16;
            Glds[m * GST + n] = acc[v];
        }
        __syncthreads();

        // LIF scan over this t-block (threads 0..63, one f each).
        // Column access Glds[tt][tid] with stride 65 -> bank (tt+tid)%64.
        if (tid < FT) {
            #pragma unroll
            for (int tt = 0; tt < TT; ++tt) {
                float g = Glds[tt * GST + tid];
                mem = 0.8f * mem + g;
                float s = (mem >= 1.0f) ? 1.0f : 0.0f;
                mem -= th * s;
                Glds[tt * GST + tid] = s;   // store spike in place
            }
        }
        __syncthreads();

        // Per-t partial reductions over the 64 f's: wave w handles 4 t rows.
        #pragma unroll
        for (int i = 0; i < 4; ++i) {
            int tt = wave * 4 + i;
            int t  = t0 + tt;
            float s0 = Glds[tt * GST + lane];
            float s1 = Glds[tt * GST + lane + 32];
            float a0 = actb[(size_t)t * F_ + f_base + lane];
            float a1 = actb[(size_t)t * F_ + f_base + lane + 32];
            float c = s0 + s1;
            float d = s0 * a0 + s1 * a1;
            float q = a0 * a0 + a1 * a1;
            #pragma unroll
            for (int off = 16; off > 0; off >>= 1) {
                c += __shfl_down(c, off, 32);
                d += __shfl_down(d, off, 32);
                q += __shfl_down(q, off, 32);
            }
            if (lane == 0) {
                cntp[pbase + t] = c;
                dotp[pbase + t] = d;
                ssp [pbase + t] = q;
            }
        }
        __syncthreads();   // Glds reused next t-block
    }
}

// Combine per-tile partials over the 64 f tiles (fixed order, deterministic)
// and form cosine per (b,t).
__global__ __launch_bounds__(256) void combine_kernel(
    const float* __restrict__ cntp, const float* __restrict__ dotp,
    const float* __restrict__ ssp, float* __restrict__ cosv)
{
    int idx = blockIdx.x * blockDim.x + threadIdx.x;
    if (idx >= B_ * T_) return;
    int b = idx / T_;
    int t = idx - b * T_;
    float c = 0.f, d = 0.f, q = 0.f;
    for (int tile = 0; tile < NT; ++tile) {
        int p = (b * NT + tile) * T_ + t;
        c += cntp[p];
        d += dotp[p];
        q += ssp[p];
    }
    float nl = sqrtf(c); if (nl < 1e-12f) nl = 1e-12f;
    float na = sqrtf(q); if (na < 1e-12f) na = 1e-12f;
    cosv[idx] = d / (na * nl);
}

// out = 1 - mean(cos); single block, fixed-order tree reduction.
__global__ __launch_bounds__(256) void final_reduce_kernel(
    const float* __restrict__ cosv, float* __restrict__ out)
{
    __shared__ float red[256];
    float s = 0.f;
    for (int i = threadIdx.x; i < B_ * T_; i += 256) s += cosv[i];
    red[threadIdx.x] = s;
    __syncthreads();
    for (int off = 128; off > 0; off >>= 1) {
        if (threadIdx.x < off) red[threadIdx.x] += red[threadIdx.x + off];
        __syncthreads();
    }
    if (threadIdx.x == 0) out[0] = 1.0f - red[0] / (float)(B_ * T_);
}

extern "C" void kernel_launch(void* const* d_in, const int* in_sizes, int n_in,
                              void* d_out, int out_size, void* d_ws, size_t ws_size,
                              hipStream_t stream) {
    const float* mlp_input      = (const float*)d_in[0];
    const float* mlp_activation = (const float*)d_in[1];
    const float* W_proj         = (const float*)d_in[2];
    const float* theta          = (const float*)d_in[3];
    float* out = (float*)d_out;

    // Workspace layout (floats): partials are fully overwritten each call.
    const size_t NP = (size_t)B_ * NT * T_;   // 524288 per quantity
    float* ws   = (float*)d_ws;
    float* cntp = ws;
    float* dotp = ws + NP;
    float* ssp  = ws + 2 * NP;
    float* cosv = ws + 3 * NP;                // 8192 floats

    lif_gemm_kernel<<<B_ * NT, 256, 0, stream>>>(
        mlp_input, mlp_activation, W_proj, theta, cntp, dotp, ssp);
    combine_kernel<<<(B_ * T_ + 255) / 256, 256, 0, stream>>>(cntp, dotp, ssp, cosv);
    final_reduce_kernel<<<1, 256, 0, stream>>>(cosv, out);
}